// QClassifier_87093346828778
// MI455X (gfx1250) — compile-verified
//
#include <hip/hip_runtime.h>

typedef float v2f __attribute__((ext_vector_type(2)));
typedef float v8f __attribute__((ext_vector_type(8)));

#define WS_UR 0
#define WS_UI 256
#define WS_M1 512

#define SWIZZLE_SWAPX16 0x401F  // group-of-32: xor=0x10, or=0x00, and=0x1f

__device__ __forceinline__ v8f wmma4(v2f a, v2f b, v8f c) {
  // D = A(16x4,f32) * B(4x16,f32) + C(16x16,f32)
  return __builtin_amdgcn_wmma_f32_16x16x4_f32(false, a, false, b, (short)0, c,
                                               false, false);
}

// K-index permutation when reusing a WMMA D accumulator as the next B operand:
// D element M bits {lh,c1,c0,v}  vs  B element K bits {c1,c0,lh,v}
__device__ __forceinline__ int permK(int kap) {
  return ((kap & 2) << 2) | ((kap & 12) >> 1) | (kap & 1);
}

__device__ __forceinline__ float xor16_add(float p) {
  int other = __builtin_amdgcn_ds_swizzle(__float_as_int(p), SWIZZLE_SWAPX16);
  return p + __int_as_float(other);
}

// ---------------------------------------------------------------------------
// Prep: build Ur, Ui (16x16 real/imag of kron(u0..u3)) and M1 = S @ w1^T.
// One block of 256 threads, one thread per (k, j) matrix element.
// ---------------------------------------------------------------------------
__global__ void qc_prep_kernel(const float* __restrict__ uw,
                               const float* __restrict__ w1,
                               float* __restrict__ ws) {
  int t = threadIdx.x;
  if (t >= 256) return;
  int k = t >> 4;
  int j = t & 15;

  // U[k][j] = prod_q u_q[bit_q(k)][bit_q(j)], wire 0 = MSB
  float ar = 1.f, ai = 0.f;
#pragma unroll
  for (int q = 0; q < 4; ++q) {
    float th = 0.5f * uw[2 * q + 0];
    float ph = 0.5f * uw[2 * q + 1];
    float c = __cosf(th), s = __sinf(th);    // native v_cos/v_sin
    float ce = __cosf(ph), se = __sinf(ph);
    int kb = (k >> (3 - q)) & 1;
    int jb = (j >> (3 - q)) & 1;
    float ry = (kb == 0) ? (jb == 0 ? c : -s) : (jb == 0 ? s : c);
    // RZ phase: row 0 -> e^{-i ph} = (ce,-se); row 1 -> e^{+i ph} = (ce,+se)
    float mr = ry * ce;
    float mi = ry * (kb ? se : -se);
    float nr = ar * mr - ai * mi;
    float ni = ar * mi + ai * mr;
    ar = nr; ai = ni;
  }
  ws[WS_UR + k * 16 + j] = ar;
  ws[WS_UI + k * 16 + j] = ai;

  // M1[k][j] = sum_f S[k][f] * w1[j][f]   (S = [Z_SIGNS | ZZ_SIGNS], 16x10)
  float z0 = 1.f - 2.f * (float)((k >> 3) & 1);
  float z1 = 1.f - 2.f * (float)((k >> 2) & 1);
  float z2 = 1.f - 2.f * (float)((k >> 1) & 1);
  float z3 = 1.f - 2.f * (float)(k & 1);
  float S[10] = {z0, z1, z2, z3, z0 * z1, z0 * z2, z0 * z3,
                 z1 * z2, z1 * z3, z2 * z3};
  float acc = 0.f;
#pragma unroll
  for (int f = 0; f < 10; ++f) acc += S[f] * w1[j * 10 + f];
  ws[WS_M1 + k * 16 + j] = acc;
}

// ---------------------------------------------------------------------------
// Main: persistent waves, 16 samples per tile, 16 f32 WMMAs per tile.
// ---------------------------------------------------------------------------
__global__ __launch_bounds__(256) void qc_main_kernel(
    const float* __restrict__ x, const float* __restrict__ b1,
    const float* __restrict__ w2, const float* __restrict__ b2,
    const float* __restrict__ ws, float* __restrict__ out, int ntiles) {
  const int lane = threadIdx.x & 31;
  const int ls = lane & 15;   // sample-in-tile (B/D lane dim)
  const int lh = lane >> 4;   // lane half (K split)

  // Wave-invariant constant operands in A layout (lane = M, K = 4c+v+2*lh).
  v2f aur[4], aui[4], am1[4], aw2[4];
  float bb[8];
#pragma unroll
  for (int c2 = 0; c2 < 4; ++c2) {
#pragma unroll
    for (int v = 0; v < 2; ++v) {
      int kap = 4 * c2 + v + 2 * lh;
      int pk = permK(kap);
      aur[c2][v] = ws[WS_UR + ls * 16 + kap];          // A1 = Ur
      aui[c2][v] = ws[WS_UI + ls * 16 + kap];          // A1' = Ui
      am1[c2][v] = ws[WS_M1 + pk * 16 + ls];           // A2[m][kap] = M1[pi(kap)][m]
      aw2[c2][v] = (ls < 2) ? w2[ls * 16 + pk] : 0.f;  // A3[m][kap] = w2[m][pi(kap)]
    }
  }
#pragma unroll
  for (int v = 0; v < 8; ++v) bb[v] = b1[v + 8 * lh];  // bias by D row m = v+8*lh
  const float b2v0 = b2[0];
  const float b2v1 = b2[1];

  const int wid = (int)((blockIdx.x * blockDim.x + threadIdx.x) >> 5);
  const int nw = (int)((gridDim.x * blockDim.x) >> 5);

  for (int tile = wid; tile < ntiles; tile += nw) {
    // ---- load X tile as B operand: lane ls = sample, j = 4c + v + 2*lh ----
    const float* xrow = x + (size_t)(tile * 16 + ls) * 16 + 2 * lh;
    if (tile + nw < ntiles)
      __builtin_prefetch(x + (size_t)((tile + nw) * 16 + ls) * 16, 0, 1);
    v2f bx[4];
#pragma unroll
    for (int c2 = 0; c2 < 4; ++c2)
      bx[c2] = __builtin_nontemporal_load((const v2f*)(xrow + 4 * c2));

    // ---- ||x||^2 per sample (halves hold complementary features) ----
    float p = 0.f;
#pragma unroll
    for (int c2 = 0; c2 < 4; ++c2)
      p += bx[c2][0] * bx[c2][0] + bx[c2][1] * bx[c2][1];
    p = xor16_add(p);                           // ds_swizzle SWAPX16
    const float recip = __builtin_amdgcn_rcpf(p);  // v_rcp_f32

    // ---- stage 1: RE = Ur * X^T, IM = Ui * X^T  (8 WMMAs) ----
    v8f re = {0, 0, 0, 0, 0, 0, 0, 0};
    v8f im = {0, 0, 0, 0, 0, 0, 0, 0};
#pragma unroll
    for (int c2 = 0; c2 < 4; ++c2) {
      re = wmma4(aur[c2], bx[c2], re);
      im = wmma4(aui[c2], bx[c2], im);
    }

    // ---- unnormalized probabilities |amp|^2 (layout: lane = sample) ----
    v8f prob;
#pragma unroll
    for (int v = 0; v < 8; ++v) prob[v] = re[v] * re[v] + im[v] * im[v];

    // ---- stage 2: H^T = relu(M1p^T * prob * recip + b1)  (4 WMMAs) ----
    // recip folded into the epilogue FMA (stage-2 output lane = sample).
    v8f h = {0, 0, 0, 0, 0, 0, 0, 0};
#pragma unroll
    for (int c2 = 0; c2 < 4; ++c2) {
      v2f pc;
      pc[0] = prob[2 * c2];
      pc[1] = prob[2 * c2 + 1];
      h = wmma4(am1[c2], pc, h);
    }
#pragma unroll
    for (int v = 0; v < 8; ++v) h[v] = fmaxf(fmaf(h[v], recip, bb[v]), 0.f);

    // ---- stage 3: OUT^T = w2p * H^T  (4 WMMAs) ----
    v8f o = {0, 0, 0, 0, 0, 0, 0, 0};
#pragma unroll
    for (int c2 = 0; c2 < 4; ++c2) {
      v2f hc;
      hc[0] = h[2 * c2];
      hc[1] = h[2 * c2 + 1];
      o = wmma4(aw2[c2], hc, o);
    }

    // ---- store: lanes 0..15 hold out[sample=lane][0..1] in o[0], o[1] ----
    if (lane < 16) {
      v2f r;
      r[0] = o[0] + b2v0;
      r[1] = o[1] + b2v1;
      __builtin_nontemporal_store(r, (v2f*)(out + (size_t)(tile * 16 + lane) * 2));
    }
  }
}

extern "C" void kernel_launch(void* const* d_in, const int* in_sizes, int n_in,
                              void* d_out, int out_size, void* d_ws,
                              size_t ws_size, hipStream_t stream) {
  const float* x = (const float*)d_in[0];
  const float* uw = (const float*)d_in[1];
  const float* w1 = (const float*)d_in[2];
  const float* b1 = (const float*)d_in[3];
  const float* w2 = (const float*)d_in[4];
  const float* b2 = (const float*)d_in[5];
  float* out = (float*)d_out;
  float* ws = (float*)d_ws;

  const int batch = in_sizes[0] / 16;
  const int ntiles = batch / 16;  // BATCH = 2^21 -> 131072 full tiles

  qc_prep_kernel<<<1, 256, 0, stream>>>(uw, w1, ws);
  qc_main_kernel<<<1024, 256, 0, stream>>>(x, b1, w2, b2, ws, out, ntiles);
}